// HamiltonianEvolution_31954556682826
// MI455X (gfx1250) — compile-verified
//
#include <hip/hip_runtime.h>
#include <stdint.h>

// ---------------------------------------------------------------------------
// HamiltonianEvolution for MI455X (gfx1250).
//
// Key identity: spectral_gate is constant along the FFT axis, so
// ifft(fft(x)*g).real == g*x. The whole op collapses to a per-channel 4x4
// linear map M_d = gate[d] * L(q_left_norm) * R(conj(q_right_norm)) applied to
// every (b,t) quaternion group. 128 MB of traffic, ~117 MFLOP -> memory-bound
// (~5.5 us at 23.3 TB/s). WMMA cannot help (256 distinct 4x4 matrices, K=4,
// any WMMA layout wrecks coalescing). CDNA5-specific paths used instead:
//   * TDM tensor_load_to_lds DMAs the 16 KB fused-matrix table into LDS
//     (s_wait_tensorcnt), zero VMEM-issue cost,
//   * global_prefetch_b8 ahead of the stream,
//   * non-temporal b128 loads/stores for the read-once/write-once stream.
// ---------------------------------------------------------------------------

#define EPSQ 1e-8f

typedef unsigned int u32x4 __attribute__((ext_vector_type(4)));
typedef int          i32x8 __attribute__((ext_vector_type(8)));
typedef int          i32x4 __attribute__((ext_vector_type(4)));
typedef float        f32x4 __attribute__((ext_vector_type(4)));

constexpr int QD  = 256;      // quat_dim = d_model/4
constexpr int DM  = 4 * QD;   // d_model = 1024
constexpr int RPB = 32;       // bt-rows per block in the streaming kernel

// ---------------------------------------------------------------------------
// Kernel 1: build the fused 4x4 matrices, one per channel d.
// Layout: M[(i*4+j)*QD + d]  (16 planes of QD floats -> float4-friendly).
// ---------------------------------------------------------------------------
__global__ void build_rot_matrix(const float* __restrict__ ql,
                                 const float* __restrict__ qr,
                                 const float* __restrict__ gate,
                                 float* __restrict__ M) {
  const int d = threadIdx.x;   // 0..QD-1
  float l[4], r[4];
  float sl = EPSQ, sr = EPSQ;
#pragma unroll
  for (int i = 0; i < 4; ++i) {
    l[i] = ql[i * QD + d];  sl = fmaf(l[i], l[i], sl);
    r[i] = qr[i * QD + d];  sr = fmaf(r[i], r[i], sr);
  }
  const float il = 1.0f / sqrtf(sl);
  const float ir = 1.0f / sqrtf(sr);
#pragma unroll
  for (int i = 0; i < 4; ++i) { l[i] *= il; r[i] *= ir; }
  // conjugate of normalized q_right
  r[1] = -r[1]; r[2] = -r[2]; r[3] = -r[3];

  const float a = l[0], b = l[1], c = l[2], dd = l[3];
  const float e = r[0], f = r[1], g = r[2], h = r[3];

  // left-multiplication matrix L(l): v -> l (x) v
  const float L[4][4] = {{a, -b, -c, -dd},
                         {b,  a, -dd, c},
                         {c,  dd, a, -b},
                         {dd, -c,  b,  a}};
  // right-multiplication matrix R(r): v -> v (x) r
  const float R[4][4] = {{e, -f, -g, -h},
                         {f,  e,  h, -g},
                         {g, -h,  e,  f},
                         {h,  g, -f,  e}};
  const float gd = gate[d];
#pragma unroll
  for (int i = 0; i < 4; ++i) {
#pragma unroll
    for (int j = 0; j < 4; ++j) {
      float s = 0.0f;
#pragma unroll
      for (int k = 0; k < 4; ++k) s = fmaf(L[i][k], R[k][j], s);
      M[(i * 4 + j) * QD + d] = gd * s;
    }
  }
}

// ---------------------------------------------------------------------------
// Kernel 2: streaming apply. One float4 column (4 channels) per thread,
// 2 bt-rows per loop iteration (8 b128 loads in flight), RPB rows per block.
// ---------------------------------------------------------------------------
__device__ inline f32x4 ldnt(const f32x4* p) { return __builtin_nontemporal_load(p); }
__device__ inline void  stnt(f32x4* p, f32x4 v) { __builtin_nontemporal_store(v, p); }

__global__ __launch_bounds__(256) void quat_apply(const float* __restrict__ x,
                                                  const float* __restrict__ Mg,
                                                  float* __restrict__ out) {
  __shared__ __align__(16) float ldsM[16 * QD];   // 16 KB matrix table

  // --- TDM: DMA the 16KB M-table from global to LDS (one wave issues it) ---
  if (threadIdx.x == 0) {
    const uint64_t ga = (uint64_t)(uintptr_t)Mg;
    const uint32_t la = (uint32_t)(uintptr_t)(&ldsM[0]);   // LDS byte offset
    const uint32_t nelem = 16u * QD;                        // 4096 x 4B

    u32x4 g0;
    g0[0] = 1u;                                   // count=1, user descriptor
    g0[1] = la;                                   // lds_addr [63:32]
    g0[2] = (uint32_t)(ga & 0xffffffffu);         // global_addr lo
    g0[3] = (uint32_t)((ga >> 32) & 0x01ffffffu)  // global_addr hi (57-bit)
          | (2u << 30);                           // type = 2 ("image")

    i32x8 g1;
    g1[0] = (int)(2u << 16);                      // data_size = 4B
    g1[1] = (int)((nelem & 0xffffu) << 16);       // tensor_dim0 [79:48] lo
    g1[2] = (int)((nelem >> 16) | (1u << 16));    // tensor_dim0 hi | tensor_dim1=1
    g1[3] = (int)((nelem & 0xffffu) << 16);       // tile_dim0 [127:112]
    g1[4] = 1;                                    // tile_dim1 = 1
    g1[5] = (int)nelem;                           // tensor_dim0_stride
    g1[6] = 0;
    g1[7] = 0;

    i32x4 gz;  gz[0] = 0; gz[1] = 0; gz[2] = 0; gz[3] = 0;
    i32x8 gz8; gz8[0] = 0; gz8[1] = 0; gz8[2] = 0; gz8[3] = 0;
               gz8[4] = 0; gz8[5] = 0; gz8[6] = 0; gz8[7] = 0;

    // 6-arg variant (clang-23 / therock-10.0 lane):
    // (group0, group1, group2, group3, group4, cpol)
    __builtin_amdgcn_tensor_load_to_lds(g0, g1, gz, gz, gz8, 0);
    __builtin_amdgcn_s_wait_tensorcnt(0);
  }
  __syncthreads();

  const int dg   = threadIdx.x & 63;   // which float4 column (4 channels)
  const int rsub = threadIdx.x >> 6;   // 0..3 row phase (uniform per wave)

  // Hoist this thread's 16 matrix float4s into registers (one-time LDS read).
  f32x4 M[4][4];
#pragma unroll
  for (int i = 0; i < 4; ++i)
#pragma unroll
    for (int j = 0; j < 4; ++j)
      M[i][j] = *reinterpret_cast<const f32x4*>(&ldsM[(i * 4 + j) * QD + dg * 4]);

  const f32x4* __restrict__ x4 = reinterpret_cast<const f32x4*>(x);
  f32x4* __restrict__ o4       = reinterpret_cast<f32x4*>(out);
  const int btBase = blockIdx.x * RPB;

  // r takes {rsub, rsub+8, rsub+16, rsub+24}; each iteration handles rows
  // r and r+4 -> 8 non-temporal b128 loads in flight per thread.
  for (int r = rsub; r < RPB; r += 8) {
    const long b0 = (long)(btBase + r) * (DM / 4) + dg;   // float4 index, row r
    const long b1 = b0 + 4 * (DM / 4);                    // row r+4

    const f32x4 a0 = ldnt(x4 + b0 + 0 * (QD / 4));
    const f32x4 a1 = ldnt(x4 + b0 + 1 * (QD / 4));
    const f32x4 a2 = ldnt(x4 + b0 + 2 * (QD / 4));
    const f32x4 a3 = ldnt(x4 + b0 + 3 * (QD / 4));
    const f32x4 c0 = ldnt(x4 + b1 + 0 * (QD / 4));
    const f32x4 c1 = ldnt(x4 + b1 + 1 * (QD / 4));
    const f32x4 c2 = ldnt(x4 + b1 + 2 * (QD / 4));
    const f32x4 c3 = ldnt(x4 + b1 + 3 * (QD / 4));

    if (r + 8 < RPB) {  // uniform branch: prefetch next iteration's rows
      __builtin_prefetch(x4 + b0 + 8 * (DM / 4), 0, 0);
      __builtin_prefetch(x4 + b1 + 8 * (DM / 4), 0, 0);
    }

#pragma unroll
    for (int i = 0; i < 4; ++i) {
      const f32x4 acc = M[i][0] * a0 + M[i][1] * a1 + M[i][2] * a2 + M[i][3] * a3;
      stnt(o4 + b0 + i * (QD / 4), acc);
    }
#pragma unroll
    for (int i = 0; i < 4; ++i) {
      const f32x4 acc = M[i][0] * c0 + M[i][1] * c1 + M[i][2] * c2 + M[i][3] * c3;
      stnt(o4 + b1 + i * (QD / 4), acc);
    }
  }
}

// ---------------------------------------------------------------------------
// Host launcher.
// Inputs (setup_inputs order): x (B*T*d_model f32), q_left (4*QD f32),
// q_right (4*QD f32), spectral_gate (QD f32). Output: B*T*d_model f32.
// ---------------------------------------------------------------------------
extern "C" void kernel_launch(void* const* d_in, const int* in_sizes, int n_in,
                              void* d_out, int out_size, void* d_ws, size_t ws_size,
                              hipStream_t stream) {
  (void)n_in; (void)out_size; (void)ws_size;
  const float* x    = (const float*)d_in[0];
  const float* ql   = (const float*)d_in[1];
  const float* qr   = (const float*)d_in[2];
  const float* gate = (const float*)d_in[3];
  float* out = (float*)d_out;
  float* Mws = (float*)d_ws;               // 16*QD floats = 16 KB scratch

  const int BT = in_sizes[0] / DM;         // 4*4096 = 16384 bt rows

  build_rot_matrix<<<1, QD, 0, stream>>>(ql, qr, gate, Mws);

  const int nblk = (BT + RPB - 1) / RPB;   // 512 blocks x 256 threads
  quat_apply<<<nblk, 256, 0, stream>>>(x, Mws, out);
}